// Gemma3Main_46583215292864
// MI455X (gfx1250) — compile-verified
//
#include <hip/hip_runtime.h>
#include <math.h>

// ---------------- Model constants ----------------
#define Tt   128
#define Dd   1152
#define NHh  4
#define HDd  256
#define Ff   6912
#define Vv   32768
#define Ll   12
#define Cc   1920
#define Ss   2048          // Cc + Tt
#define QKVd 1536          // NG*(NH/NG+2)*HD
#define CAPf 50.0f
#define EPSf 1e-6f

typedef float v2f __attribute__((ext_vector_type(2)));
typedef float v8f __attribute__((ext_vector_type(8)));

// ---------------- WMMA fp32 GEMM (A given K-major) ----------------
// Computes C[z][128, N] = A[z][128, K] * B[z][K, N] where A is supplied
// TRANSPOSED (AT: K x 128, K-major) so fragment loads are lane-coalesced.
// TRANSC=false: C stored (128, N) row-major.
// TRANSC=true : C stored (N, 128) row-major (K-major for the next GEMM);
//               per-lane the 8 acc elements are 8 consecutive M values ->
//               contiguous 32B stores.
// One block = 4 waves; each wave does a 16-wide N stripe over all 8 M-tiles,
// so each B fragment feeds 8 independent v_wmma_f32_16x16x4_f32.
template <bool TRANSC>
__global__ __launch_bounds__(128) void gemm128t_wmma(
    const float* __restrict__ AT, const float* __restrict__ B,
    float* __restrict__ C, int N, int K,
    int ldat, int ldb, int ldc,
    long strideA, long strideB, long strideC)
{
    const int wave = threadIdx.x >> 5;
    const int lane = threadIdx.x & 31;
    const int n0 = (blockIdx.x * 4 + wave) * 16;
    if (n0 >= N) return;
    const int z = blockIdx.z;
    AT += (long)z * strideA;
    B  += (long)z * strideB;
    C  += (long)z * strideC;
    const int r    = lane & 15;   // m (A frag) / n (B,C frag) within tile
    const int half = lane >> 4;   // selects k-pair within the 4-wide K step

    v8f acc[8];
#pragma unroll
    for (int i = 0; i < 8; ++i) acc[i] = (v8f){0, 0, 0, 0, 0, 0, 0, 0};

    for (int k = 0; k < K; k += 4) {
        const int ka = k + 2 * half;
        v2f b;
        b.x = B[(long)ka * ldb + n0 + r];
        b.y = B[(long)(ka + 1) * ldb + n0 + r];
        const float* a0 = AT + (long)ka * ldat + r;
        const float* a1 = a0 + ldat;
#pragma unroll
        for (int mi = 0; mi < 8; ++mi) {
            v2f a;
            a.x = a0[mi * 16];
            a.y = a1[mi * 16];
            acc[mi] = __builtin_amdgcn_wmma_f32_16x16x4_f32(
                false, a, false, b, (short)0, acc[mi], false, false);
        }
    }
    if (TRANSC) {
#pragma unroll
        for (int mi = 0; mi < 8; ++mi) {
            float* cp = C + (long)(n0 + r) * ldc + mi * 16 + 8 * half;
#pragma unroll
            for (int j = 0; j < 8; ++j) cp[j] = acc[mi][j];
        }
    } else {
#pragma unroll
        for (int mi = 0; mi < 8; ++mi) {
#pragma unroll
            for (int j = 0; j < 8; ++j) {
                C[(long)(mi * 16 + j + 8 * half) * ldc + n0 + r] = acc[mi][j];
            }
        }
    }
}

// ---------------- LDS-tiled batched transpose ----------------
// src: (M, N) leading dim sld ; dst: (N, M) leading dim dld.
// Assumes M, N multiples of 32. grid = (N/32, M/32, batch), block = (32, 8).
__global__ __launch_bounds__(256) void transpose_kernel(
    const float* __restrict__ src, float* __restrict__ dst,
    int sld, int dld, long szs, long dzs)
{
    __shared__ float tile[32][33];
    src += (long)blockIdx.z * szs;
    dst += (long)blockIdx.z * dzs;
    const int bx = blockIdx.x * 32;   // src col base
    const int by = blockIdx.y * 32;   // src row base
    const int tx = threadIdx.x, ty = threadIdx.y;
#pragma unroll
    for (int i = 0; i < 32; i += 8)
        tile[ty + i][tx] = src[(long)(by + ty + i) * sld + bx + tx];
    __syncthreads();
#pragma unroll
    for (int i = 0; i < 32; i += 8)
        dst[(long)(bx + ty + i) * dld + by + tx] = tile[tx][ty + i];
}

// ---------------- RMSNorm (optional fused residual add) ----------------
__global__ __launch_bounds__(256) void rms_kernel(
    const float* __restrict__ x, const float* __restrict__ w,
    const float* __restrict__ base, float* __restrict__ out, int dim)
{
    __shared__ float red[256];
    const int row = blockIdx.x;
    const float* xr = x + (long)row * dim;
    float s = 0.f;
    for (int j = threadIdx.x; j < dim; j += 256) {
        float v = xr[j];
        s += v * v;
    }
    red[threadIdx.x] = s;
    __syncthreads();
    for (int o = 128; o > 0; o >>= 1) {
        if (threadIdx.x < o) red[threadIdx.x] += red[threadIdx.x + o];
        __syncthreads();
    }
    const float rs = rsqrtf(red[0] / (float)dim + EPSf);
    float* orow = out + (long)row * dim;
    const float* brow = base ? base + (long)row * dim : nullptr;
    for (int j = threadIdx.x; j < dim; j += 256) {
        float v = xr[j] * rs * (1.f + w[j]);
        orow[j] = brow ? brow[j] + v : v;
    }
}

// ---------------- QKV post: per-head RMS + RoPE, scatter K/V ----------------
// qkvT layout: (QKV, T) K-major. slots along QKV: [q0 q1 q2 q3 k v] x 256.
// One block per token t; thread d = 0..255.
__global__ __launch_bounds__(256) void qkv_post_kernel(
    const float* __restrict__ qkvT,
    const float* __restrict__ qw, const float* __restrict__ kw,
    const float* __restrict__ cs, const float* __restrict__ sn,
    float* __restrict__ qT,     // (NH*HD, T)
    float* __restrict__ kfT,    // (HD, S)
    float* __restrict__ vfT,    // (S, HD)
    float* __restrict__ kslice, // (T, HD)
    float* __restrict__ vslice) // (HD, T)
{
    __shared__ float red[256];
    __shared__ float buf[256];
    const int t = blockIdx.x, d = threadIdx.x;
    const float c = cs[t * HDd + d];
    const float s = sn[t * HDd + d];

    for (int slot = 0; slot < 5; ++slot) {
        const float x = qkvT[(long)(slot * HDd + d) * Tt + t];
        red[d] = x * x;
        __syncthreads();
        for (int o = 128; o > 0; o >>= 1) {
            if (d < o) red[d] += red[d + o];
            __syncthreads();
        }
        const float rs = rsqrtf(red[0] / (float)HDd + EPSf);
        const float* w = (slot < 4) ? qw : kw;
        const float n = x * rs * (1.f + w[d]);
        buf[d] = n;
        __syncthreads();
        const float rot = (d < HDd / 2) ? -buf[d + HDd / 2] : buf[d - HDd / 2];
        const float o = n * c + rot * s;
        if (slot < 4) {
            qT[(long)(slot * HDd + d) * Tt + t] = o;
        } else {
            kfT[(long)d * Ss + Cc + t] = o;
            kslice[(long)t * HDd + d] = o;
        }
        __syncthreads();
    }
    const float v = qkvT[(long)(5 * HDd + d) * Tt + t];
    vfT[(long)(Cc + t) * HDd + d] = v;
    vslice[(long)d * Tt + t] = v;
}

// ---------------- tanh-capped masked softmax over S=2048 ----------------
__global__ __launch_bounds__(256) void softmax_cap_kernel(
    float* __restrict__ att, const float* __restrict__ mask, float scale)
{
    __shared__ float red[256];
    const int h = blockIdx.y, t = blockIdx.x;
    float* rowp = att + ((long)h * Tt + t) * Ss;
    const float* mrow = mask + (long)t * Ss;
    float vals[8];
    float mx = -1e30f;
#pragma unroll
    for (int i = 0; i < 8; ++i) {
        const int s = threadIdx.x + i * 256;
        float v = rowp[s] * scale;
        v = tanhf(v * (1.f / CAPf)) * CAPf + mrow[s];
        vals[i] = v;
        mx = fmaxf(mx, v);
    }
    red[threadIdx.x] = mx;
    __syncthreads();
    for (int o = 128; o > 0; o >>= 1) {
        if (threadIdx.x < o) red[threadIdx.x] = fmaxf(red[threadIdx.x], red[threadIdx.x + o]);
        __syncthreads();
    }
    mx = red[0];
    __syncthreads();
    float sum = 0.f;
#pragma unroll
    for (int i = 0; i < 8; ++i) {
        vals[i] = expf(vals[i] - mx);
        sum += vals[i];
    }
    red[threadIdx.x] = sum;
    __syncthreads();
    for (int o = 128; o > 0; o >>= 1) {
        if (threadIdx.x < o) red[threadIdx.x] += red[threadIdx.x + o];
        __syncthreads();
    }
    const float inv = 1.f / red[0];
#pragma unroll
    for (int i = 0; i < 8; ++i) rowp[threadIdx.x + i * 256] = vals[i] * inv;
}

// ---------------- GELU(tanh) * up (layout agnostic elementwise) ----------------
__global__ __launch_bounds__(256) void gelu_mul_kernel(
    const float* __restrict__ g, const float* __restrict__ u,
    float* __restrict__ o, int n)
{
    const int i = blockIdx.x * 256 + threadIdx.x;
    if (i < n) {
        const float x = g[i];
        const float inner = 0.7978845608028654f * (x + 0.044715f * x * x * x);
        o[i] = 0.5f * x * (1.f + tanhf(inner)) * u[i];
    }
}

// ---------------- workspace layout (floats) ----------------
static const long OFF_H    = 0;        // 147456 (T,D)
static const long OFF_X    = 147456;   // 147456 (T,D)
static const long OFF_XT   = 294912;   // 147456 (D,T)
static const long OFF_QKVT = 442368;   // 196608 (QKV,T)
static const long OFF_QT   = 638976;   // 131072 (NH*HD,T)
static const long OFF_KFT  = 770048;   // 524288 (HD,S)
static const long OFF_VFT  = 1294336;  // 524288 (S,HD)
static const long OFF_ATT  = 1818624;  // 1048576 (NH,T,S)
static const long OFF_ATTT = 2867200;  // 1048576 (NH,S,T)
static const long OFF_AOT  = 3915776;  // 131072 (NH*HD,T)
static const long OFF_PROJ = 4046848;  // 147456 (T,D)
static const long OFF_GT   = 4194304;  // 884736 (F,T)
static const long OFF_UT   = 5079040;  // 884736 (F,T)
static const long OFF_FF   = 5963776;  // 147456 (T,D)  -> total 6111232 f (~23.3 MB)

extern "C" void kernel_launch(void* const* d_in, const int* in_sizes, int n_in,
                              void* d_out, int out_size, void* d_ws, size_t ws_size,
                              hipStream_t stream)
{
    const float* embeddings  = (const float*)d_in[0];
    const float* mask_global = (const float*)d_in[1];
    const float* mask_local  = (const float*)d_in[2];
    const float* pe_cos      = (const float*)d_in[3];
    const float* pe_sin      = (const float*)d_in[4];
    const float* pe_cos_loc  = (const float*)d_in[5];
    const float* pe_sin_loc  = (const float*)d_in[6];
    const float* kv_k        = (const float*)d_in[7];
    const float* kv_v        = (const float*)d_in[8];
    const float* pre_attn_w  = (const float*)d_in[9];
    const float* q_norm_w    = (const float*)d_in[10];
    const float* k_norm_w    = (const float*)d_in[11];
    const float* post_attn_w = (const float*)d_in[12];
    const float* pre_ff_w    = (const float*)d_in[13];
    const float* post_ff_w   = (const float*)d_in[14];
    const float* final_w     = (const float*)d_in[15];
    const float* w_qkv       = (const float*)d_in[16];
    const float* w_out       = (const float*)d_in[17];
    const float* w_gate      = (const float*)d_in[18];
    const float* w_up        = (const float*)d_in[19];
    const float* w_down      = (const float*)d_in[20];
    const float* w_lm        = (const float*)d_in[21];

    float* ws    = (float*)d_ws;
    float* h     = ws + OFF_H;
    float* x     = ws + OFF_X;
    float* xT    = ws + OFF_XT;
    float* qkvT  = ws + OFF_QKVT;
    float* qT    = ws + OFF_QT;
    float* kfT   = ws + OFF_KFT;
    float* vfT   = ws + OFF_VFT;
    float* attn  = ws + OFF_ATT;
    float* attnT = ws + OFF_ATTT;
    float* aoT   = ws + OFF_AOT;
    float* proj  = ws + OFF_PROJ;
    float* gT    = ws + OFF_GT;
    float* uT    = ws + OFF_UT;
    float* ffo   = ws + OFF_FF;

    float* logits  = (float*)d_out;                   // T*V
    float* kslices = logits + (long)Tt * Vv;          // L*T*HD
    float* vslices = kslices + (long)Ll * Tt * HDd;   // L*HD*T

    const float scale = 1.f / sqrtf((float)HDd);
    const dim3 tb(32, 8);

    // h = embeddings
    hipMemcpyAsync(h, embeddings, (size_t)Tt * Dd * sizeof(float),
                   hipMemcpyDeviceToDevice, stream);

    for (int i = 0; i < Ll; ++i) {
        const bool is_local = ((i + 1) % 6) == 0;
        const float* cs = is_local ? pe_cos_loc : pe_cos;
        const float* sn = is_local ? pe_sin_loc : pe_sin;
        const float* mk = is_local ? mask_local : mask_global;

        // x = rms(h, pre_attn); xT = x^T
        rms_kernel<<<Tt, 256, 0, stream>>>(h, pre_attn_w + (long)i * Dd, nullptr, x, Dd);
        transpose_kernel<<<dim3(Dd / 32, Tt / 32, 1), tb, 0, stream>>>(x, xT, Dd, Tt, 0, 0);

        // qkvT = (x @ w_qkv[i])^T   stored (QKV, T)
        gemm128t_wmma<true><<<dim3(QKVd / 64, 1, 1), 128, 0, stream>>>(
            xT, w_qkv + (long)i * Dd * QKVd, qkvT, QKVd, Dd, Tt, QKVd, Tt, 0, 0, 0);

        // KV caches, transposed into GEMM-friendly layouts:
        //   kfT (HD, S) <- kv_k[i] (C, HD);  vfT (S, HD) <- kv_v[i] (HD, C)
        transpose_kernel<<<dim3(HDd / 32, Cc / 32, 1), tb, 0, stream>>>(
            kv_k + (long)i * Cc * HDd, kfT, HDd, Ss, 0, 0);
        transpose_kernel<<<dim3(Cc / 32, HDd / 32, 1), tb, 0, stream>>>(
            kv_v + (long)i * HDd * Cc, vfT, Cc, HDd, 0, 0);

        // per-head rms + rope; fills qT, tails of kfT/vfT, k/v slice outputs
        qkv_post_kernel<<<Tt, 256, 0, stream>>>(
            qkvT, q_norm_w + (long)i * HDd, k_norm_w + (long)i * HDd, cs, sn,
            qT, kfT, vfT, kslices + (long)i * Tt * HDd, vslices + (long)i * HDd * Tt);

        // scores[h] = q[h] @ k^T : NN with AT=qT (HD,T per head), B=kfT (HD,S)
        gemm128t_wmma<false><<<dim3(Ss / 64, 1, NHh), 128, 0, stream>>>(
            qT, kfT, attn, Ss, HDd, Tt, Ss, Ss,
            (long)HDd * Tt, 0, (long)Tt * Ss);

        // softmax(tanh-cap + mask), then transpose rows->K-major per head
        softmax_cap_kernel<<<dim3(Tt, NHh), 256, 0, stream>>>(attn, mk, scale);
        transpose_kernel<<<dim3(Ss / 32, Tt / 32, NHh), tb, 0, stream>>>(
            attn, attnT, Ss, Tt, (long)Tt * Ss, (long)Ss * Tt);

        // out[h] = attn[h] @ v^T : NN with AT=attnT (S,T per head), B=vfT (S,HD)
        // TRANSC -> aoT stored (NH*HD, T): ready as AT for out-proj.
        gemm128t_wmma<true><<<dim3(HDd / 64, 1, NHh), 128, 0, stream>>>(
            attnT, vfT, aoT, HDd, Ss, Tt, HDd, Tt,
            (long)Ss * Tt, 0, (long)HDd * Tt);

        // proj = ao @ w_out[i]
        gemm128t_wmma<false><<<dim3(Dd / 64, 1, 1), 128, 0, stream>>>(
            aoT, w_out + (long)i * NHh * HDd * Dd, proj, Dd, NHh * HDd,
            Tt, Dd, Dd, 0, 0, 0);

        // h += rms(proj, post_attn)
        rms_kernel<<<Tt, 256, 0, stream>>>(proj, post_attn_w + (long)i * Dd, h, h, Dd);

        // FFN: x = rms(h); xT; gate/up emitted K-major; gelu*up; down; h += rms
        rms_kernel<<<Tt, 256, 0, stream>>>(h, pre_ff_w + (long)i * Dd, nullptr, x, Dd);
        transpose_kernel<<<dim3(Dd / 32, Tt / 32, 1), tb, 0, stream>>>(x, xT, Dd, Tt, 0, 0);
        gemm128t_wmma<true><<<dim3(Ff / 64, 1, 1), 128, 0, stream>>>(
            xT, w_gate + (long)i * Dd * Ff, gT, Ff, Dd, Tt, Ff, Tt, 0, 0, 0);
        gemm128t_wmma<true><<<dim3(Ff / 64, 1, 1), 128, 0, stream>>>(
            xT, w_up + (long)i * Dd * Ff, uT, Ff, Dd, Tt, Ff, Tt, 0, 0, 0);
        gelu_mul_kernel<<<(Tt * Ff + 255) / 256, 256, 0, stream>>>(gT, uT, gT, Tt * Ff);
        gemm128t_wmma<false><<<dim3(Dd / 64, 1, 1), 128, 0, stream>>>(
            gT, w_down + (long)i * Ff * Dd, ffo, Dd, Ff, Tt, Dd, Dd, 0, 0, 0);
        rms_kernel<<<Tt, 256, 0, stream>>>(ffo, post_ff_w + (long)i * Dd, h, h, Dd);
    }

    // logits = rms(h, final) @ w_lm
    rms_kernel<<<Tt, 256, 0, stream>>>(h, final_w, nullptr, x, Dd);
    transpose_kernel<<<dim3(Dd / 32, Tt / 32, 1), tb, 0, stream>>>(x, xT, Dd, Tt, 0, 0);
    gemm128t_wmma<false><<<dim3(Vv / 64, 1, 1), 128, 0, stream>>>(
        xT, w_lm, logits, Vv, Dd, Tt, Vv, Vv, 0, 0, 0);
}